// CNEncoder_32658931319277
// MI455X (gfx1250) — compile-verified
//
#include <hip/hip_runtime.h>
#include <hip/hip_bf16.h>
#include <math.h>

// Problem constants (from reference)
#define NN 20000
#define GG 3000
#define EE 120000
#define EPSF 1e-8f

typedef float v2f __attribute__((ext_vector_type(2)));
typedef float v8f __attribute__((ext_vector_type(8)));
typedef float f4  __attribute__((ext_vector_type(4)));

__device__ __forceinline__ unsigned pcg_hash(unsigned x) {
  unsigned s = x * 747796405u + 2891336453u;
  unsigned w = ((s >> ((s >> 28u) + 4u)) ^ s) * 277803737u;
  return (w >> 22u) ^ w;
}

// ---------------------------------------------------------------------------
// Pass A: fused streaming pass over rf and norm_x (480 MB read, the dominant
// cost: ~20us of HBM time at 23.3 TB/s). Each wave owns a 16-row tile.
// V_WMMA_F32_16X16X4_F32 with all-ones B computes D[m,n] = sum_k A[m,k]:
// exact f32 row sums, 16 rows x 4 cols per wmma. A layout (ISA 7.12.2,
// 32-bit A 16x4): lane l<16 holds A[l&15,{0,1}], lane l>=16 holds A[l&15,{2,3}]
// -> each lane streams a float2 with stride 4. Unroll 6 (750 = 6*125 exact)
// keeps 12 b64 loads in flight per wave for memory-level parallelism; NT
// hints keep the one-shot stream from churning the 192 MB L2.
// Side products: per-row min/max of rf (closed-form global min/max later)
// and global sum(rf^2) (reg_loss).
// ---------------------------------------------------------------------------
__global__ void __launch_bounds__(256)
pass_a_kernel(const float* __restrict__ rf, const float* __restrict__ nx,
              float* __restrict__ rowsum, float* __restrict__ pseudo,
              float* __restrict__ rowmin, float* __restrict__ rowmax,
              float* __restrict__ sumsq) {
  const int wave = (int)((blockIdx.x * blockDim.x + threadIdx.x) >> 5);
  const int lane = (int)(threadIdx.x & 31u);
  if (wave >= (NN / 16)) return;          // uniform per wave: EXEC all-ones inside
  const int r    = lane & 15;
  const int half = lane >> 4;
  const int row  = wave * 16 + r;
  const float* prf = rf + (size_t)row * GG + 2 * half;
  const float* pnx = nx + (size_t)row * GG + 2 * half;

  v8f crf = {0.f, 0.f, 0.f, 0.f, 0.f, 0.f, 0.f, 0.f};
  v8f cnx = crf;
  const v2f ones = {1.0f, 1.0f};
  float lmin = 3.402823466e38f, lmax = -3.402823466e38f, ss = 0.0f;

  #pragma unroll 6
  for (int k = 0; k < GG; k += 4) {
    v2f a = __builtin_nontemporal_load((const v2f*)(prf + k));
    v2f b = __builtin_nontemporal_load((const v2f*)(pnx + k));
    crf = __builtin_amdgcn_wmma_f32_16x16x4_f32(false, a, false, ones, (short)0, crf, false, false);
    cnx = __builtin_amdgcn_wmma_f32_16x16x4_f32(false, b, false, ones, (short)0, cnx, false, false);
    lmin = fminf(lmin, fminf(a.x, a.y));
    lmax = fmaxf(lmax, fmaxf(a.x, a.y));
    ss += a.x * a.x + a.y * a.y;
  }

  // combine the two half-row streams (lane l and lane l+16 cover one row)
  float omin = __shfl_xor(lmin, 16, 32);
  float omax = __shfl_xor(lmax, 16, 32);
  lmin = fminf(lmin, omin);
  lmax = fmaxf(lmax, omax);
  if (half == 0) { rowmin[row] = lmin; rowmax[row] = lmax; }

  // C/D layout: lane n<16 VGPR v = D[v,n]; lane n+16 VGPR v = D[v+8,n].
  // Column 0 of each half (lanes 0 and 16) carries the 16 row sums.
  if (r == 0) {
    const int base = wave * 16 + 8 * half;
    #pragma unroll
    for (int v = 0; v < 8; ++v) { rowsum[base + v] = crf[v]; pseudo[base + v] = cnx[v]; }
  }

  #pragma unroll
  for (int off = 16; off > 0; off >>= 1) ss += __shfl_xor(ss, off, 32);
  if (lane == 0) atomicAdd(sumsq, ss);
}

// ---------------------------------------------------------------------------
// Single block (1024 threads = 32 wave32 waves): mean/std of per-spot means,
// then the MCMC likelihoods L[s] = sum_j exp(-0.5*(spot_mean_j - m[s])^2)
// (m[s] = raw mean of spot s, faithful quirk of the reference).
// ---------------------------------------------------------------------------
__global__ void __launch_bounds__(1024)
stats_kernel(const float* __restrict__ rowsum, float* __restrict__ scal) {
  __shared__ float sa[1024], sb[1024], sc[1024];
  __shared__ float sh_mean, sh_inv;
  const int t = threadIdx.x;
  const float invG = 1.0f / (float)GG;

  float s1 = 0.f, s2 = 0.f;
  for (int i = t; i < NN; i += 1024) {
    float m = rowsum[i] * invG;
    s1 += m; s2 += m * m;
  }
  sa[t] = s1; sb[t] = s2;
  __syncthreads();
  for (int off = 512; off > 0; off >>= 1) {
    if (t < off) { sa[t] += sa[t + off]; sb[t] += sb[t + off]; }
    __syncthreads();
  }
  if (t == 0) {
    float mean = sa[0] / (float)NN;
    float var  = fmaxf(sb[0] / (float)NN - mean * mean, 0.0f);
    float sd   = sqrtf(var);
    sh_mean = mean;
    sh_inv  = 1.0f / (sd + EPSF);
    scal[2] = mean; scal[3] = sd;
  }
  __syncthreads();

  const float mean = sh_mean, inv = sh_inv;
  const float m0 = rowsum[0] * invG, m1 = rowsum[1] * invG, m2 = rowsum[2] * invG;
  float l0 = 0.f, l1 = 0.f, l2 = 0.f;
  for (int i = t; i < NN; i += 1024) {
    float sm = (rowsum[i] * invG - mean) * inv;
    float d0 = sm - m0, d1 = sm - m1, d2 = sm - m2;
    l0 += expf(-0.5f * d0 * d0);
    l1 += expf(-0.5f * d1 * d1);
    l2 += expf(-0.5f * d2 * d2);
  }
  sa[t] = l0; sb[t] = l1; sc[t] = l2;
  __syncthreads();
  for (int off = 512; off > 0; off >>= 1) {
    if (t < off) { sa[t] += sa[t + off]; sb[t] += sb[t + off]; sc[t] += sc[t + off]; }
    __syncthreads();
  }
  if (t == 0) { scal[4] = sa[0]; scal[5] = sb[0]; scal[6] = sc[0]; }
}

// ---------------------------------------------------------------------------
// Metropolis-Hastings, 20 iterations per spot. Deterministic counter-based
// PRNG (JAX threefry bit-parity is not reproducible here; structure is
// faithful). TRANS is symmetric -> transition ratio == 1.
// ---------------------------------------------------------------------------
__global__ void __launch_bounds__(256)
mcmc_kernel(const float* __restrict__ scal, float* __restrict__ sp) {
  const int n = blockIdx.x * blockDim.x + threadIdx.x;
  if (n >= NN) return;
  const float Ls[3]  = { scal[4], scal[5], scal[6] };
  const float PRI[3] = { 0.1f, 0.8f, 0.1f };
  unsigned s = pcg_hash(0x243F6A88u ^ (unsigned)n) % 3u;
  float c0 = 0.f, c1 = 0.f, c2 = 0.f;
  #pragma unroll
  for (int it = 0; it < 20; ++it) {
    unsigned h1 = pcg_hash((unsigned)n * 1664525u + (unsigned)it * 1013904223u + 0xB5297A4Du);
    unsigned h2 = pcg_hash(h1 ^ 0x68E31DA4u);
    unsigned cand = h1 % 3u;
    float acc = (PRI[cand] / PRI[s]) * (Ls[cand] / Ls[s]);
    float u = (float)(h2 >> 8) * (1.0f / 16777216.0f);
    if (u < acc) s = cand;
    c0 += (s == 0u) ? 1.f : 0.f;
    c1 += (s == 1u) ? 1.f : 0.f;
    c2 += (s == 2u) ? 1.f : 0.f;
  }
  sp[n * 3 + 0] = c0; sp[n * 3 + 1] = c1; sp[n * 3 + 2] = c2;
}

// scatter_mean numerator/denominator over the spatial graph (f32 atomics)
__global__ void __launch_bounds__(256)
scatter_kernel(const int* __restrict__ ei, const float* __restrict__ sp,
               float* __restrict__ sums, float* __restrict__ cnt) {
  const int e = blockIdx.x * blockDim.x + threadIdx.x;
  if (e >= EE) return;
  const int r = ei[e];
  const int c = ei[EE + e];
  atomicAdd(&sums[c * 3 + 0], sp[r * 3 + 0]);
  atomicAdd(&sums[c * 3 + 1], sp[r * 3 + 1]);
  atomicAdd(&sums[c * 3 + 2], sp[r * 3 + 2]);
  atomicAdd(&cnt[c], 1.0f);
}

// ---------------------------------------------------------------------------
// Per-spot finalize + closed-form global rescale. Since c_i >= 0:
//   gmin = min_i c_i*rowmin_i, gmax = max_i c_i*rowmax_i,
//   mean_x = sum_i c_i*rowsum_i / (N*G)
// The whole min/max/mean normalization is affine: out = rf*alpha_i + beta.
// ---------------------------------------------------------------------------
__global__ void __launch_bounds__(1024)
finalize_kernel(const float* __restrict__ rowsum, const float* __restrict__ pseudo,
                const float* __restrict__ rowmin, const float* __restrict__ rowmax,
                const float* __restrict__ sums, const float* __restrict__ cnt,
                float* __restrict__ alpha, float* __restrict__ scal,
                float* __restrict__ reg_out) {
  __shared__ float smin[1024], smax[1024], ssum[1024];
  __shared__ float sh_a, sh_b;
  const int t = threadIdx.x;
  float gmin = 3.402823466e38f, gmax = -3.402823466e38f, sx = 0.f;
  for (int i = t; i < NN; i += 1024) {
    float ic = 1.0f / fmaxf(cnt[i], 1.0f);
    float s0 = sums[i * 3 + 0] * ic, s1 = sums[i * 3 + 1] * ic, s2 = sums[i * 3 + 2] * ic;
    int am = 0; float best = s0;
    if (s1 > best) { best = s1; am = 1; }
    if (s2 > best) { best = s2; am = 2; }
    float st = (float)(am + 1);
    float rs = rowsum[i];
    float ci = (st * pseudo[i]) / (st * rs + EPSF);   // norm_copy row scale
    alpha[i] = ci;                                    // provisional; scaled below
    gmin = fminf(gmin, ci * rowmin[i]);
    gmax = fmaxf(gmax, ci * rowmax[i]);
    sx += ci * rs;
  }
  smin[t] = gmin; smax[t] = gmax; ssum[t] = sx;
  __syncthreads();
  for (int off = 512; off > 0; off >>= 1) {
    if (t < off) {
      smin[t] = fminf(smin[t], smin[t + off]);
      smax[t] = fmaxf(smax[t], smax[t + off]);
      ssum[t] += ssum[t + off];
    }
    __syncthreads();
  }
  if (t == 0) {
    float gmn = smin[0], gmx = smax[0];
    float rmn = gmn * 0.8f, rmx = gmx * 1.2f;
    float scale = (rmx - rmn) / (gmx - gmn + EPSF);
    float mean_x = ssum[0] / ((float)NN * (float)GG);
    float mean_t = (mean_x - gmn) * scale + rmn;      // mean after affine rescale
    float invmt = 1.0f / mean_t;
    sh_a = scale * invmt;
    sh_b = (rmn - gmn * scale) * invmt;
    scal[1] = sh_b;
    reg_out[0] = scal[0] * 1e-4f;                     // reg_loss
  }
  __syncthreads();
  const float a = sh_a;
  for (int i = t; i < NN; i += 1024) alpha[i] *= a;
}

// ---------------------------------------------------------------------------
// Final streaming pass: 240 MB read + 240 MB write, pure FMA. Non-temporal
// hints keep the 480 MB stream from churning L2; prefetch runs ahead.
// ---------------------------------------------------------------------------
__global__ void __launch_bounds__(256)
final_elem_kernel(const float* __restrict__ rf, const float* __restrict__ alpha,
                  const float* __restrict__ scal, float* __restrict__ out) {
  const unsigned total4 = (unsigned)((size_t)NN * GG / 4);   // 15,000,000
  const float beta = scal[1];
  const f4* rf4 = (const f4*)rf;
  f4* out4 = (f4*)out;
  const unsigned stride = gridDim.x * blockDim.x;
  for (unsigned idx = blockIdx.x * blockDim.x + threadIdx.x; idx < total4; idx += stride) {
    unsigned row = idx / 750u;                               // GG/4 vec4 per row
    float a = alpha[row];
    __builtin_prefetch(rf4 + idx + stride, 0, 0);            // global_prefetch_b8
    f4 v = __builtin_nontemporal_load(rf4 + idx);
    f4 o;
    o.x = fmaf(v.x, a, beta);
    o.y = fmaf(v.y, a, beta);
    o.z = fmaf(v.z, a, beta);
    o.w = fmaf(v.w, a, beta);
    __builtin_nontemporal_store(o, out4 + idx);
  }
}

__global__ void __launch_bounds__(256)
zero_ws_kernel(float* __restrict__ ws, int n) {
  int i = blockIdx.x * blockDim.x + threadIdx.x;
  if (i < n) ws[i] = 0.0f;
}

extern "C" void kernel_launch(void* const* d_in, const int* in_sizes, int n_in,
                              void* d_out, int out_size, void* d_ws, size_t ws_size,
                              hipStream_t stream) {
  const float* nx = (const float*)d_in[0];   // norm_x (N,G)
  const float* rf = (const float*)d_in[1];   // reconstructed_features (N,G)
  const int*   ei = (const int*)d_in[2];     // edge_index (2,E)
  float* out = (float*)d_out;                // norm_copy (N*G) ++ reg_loss (1)
  float* ws  = (float*)d_ws;

  // Workspace layout (floats): 12*NN + 16 = 240,016 (~0.96 MB)
  float* rowsum = ws;            // N : sum_j rf[i,j]
  float* pseudo = ws + 1 * NN;   // N : sum_j norm_x[i,j]
  float* rowmin = ws + 2 * NN;   // N
  float* rowmax = ws + 3 * NN;   // N
  float* sp     = ws + 4 * NN;   // N*3 : MCMC visit counts
  float* sums   = ws + 7 * NN;   // N*3 : scatter numerators
  float* cnt    = ws + 10 * NN;  // N   : scatter denominators
  float* alpha  = ws + 11 * NN;  // N   : final per-row scale
  float* scal   = ws + 12 * NN;  // [0]=sumsq [1]=beta [2]=mean [3]=sd [4..6]=L
  const int ws_floats = 12 * NN + 16;

  zero_ws_kernel<<<(ws_floats + 255) / 256, 256, 0, stream>>>(ws, ws_floats);
  pass_a_kernel<<<(NN / 16 + 7) / 8, 256, 0, stream>>>(rf, nx, rowsum, pseudo,
                                                       rowmin, rowmax, &scal[0]);
  stats_kernel<<<1, 1024, 0, stream>>>(rowsum, scal);
  mcmc_kernel<<<(NN + 255) / 256, 256, 0, stream>>>(scal, sp);
  scatter_kernel<<<(EE + 255) / 256, 256, 0, stream>>>(ei, sp, sums, cnt);
  finalize_kernel<<<1, 1024, 0, stream>>>(rowsum, pseudo, rowmin, rowmax, sums, cnt,
                                          alpha, scal, out + (size_t)NN * GG);
  final_elem_kernel<<<2048, 256, 0, stream>>>(rf, alpha, scal, out);
}